// Quantize_12670153523895
// MI455X (gfx1250) — compile-verified
//
#include <hip/hip_runtime.h>

typedef __attribute__((ext_vector_type(2))) float v2f;
typedef __attribute__((ext_vector_type(8))) float v8f;

#define DIM      64
#define NEMBED   2048
#define NTOTAL   4194304            // 16*64*64*64 elements of x
#define NROWS    (NTOTAL / DIM)     // 65536 vectors
#define DIFF_OFF NTOTAL
#define IND_OFF  (NTOTAL + 1)
#define LDSP     68                 // codeword-major LDS stride (dwords):
                                    // b64 reads land on banks 4*(m+k)+2*half -> conflict-free
#define NSTAGES  (NEMBED / 64)      // 32 stages of 64 codewords

#if __has_builtin(__builtin_amdgcn_global_load_async_to_lds_b32)
#define HAVE_ASYNC_LDS 1
typedef __attribute__((address_space(1))) int gas_i;   // global int*
typedef __attribute__((address_space(3))) int las_i;   // LDS int*
#else
#define HAVE_ASYNC_LDS 0
#endif

// ---------------------------------------------------------------------------
// Pass 1: per-codeword squared norms into workspace; zero the diff accumulator.
// ---------------------------------------------------------------------------
__global__ __launch_bounds__(256) void vq_enorm_kernel(
    const float* __restrict__ embed, float* __restrict__ enorm,
    float* __restrict__ diff_slot) {
  const int n = blockIdx.x * 256 + threadIdx.x;
  if (n == 0) *diff_slot = 0.0f;
  if (n < NEMBED) {
    float s = 0.0f;
#pragma unroll
    for (int d = 0; d < DIM; ++d) {
      float e = embed[d * NEMBED + n];
      s += e * e;
    }
    enorm[n] = s;
  }
}

// ---------------------------------------------------------------------------
// Pass 2: fused distance-GEMM (WMMA f32 16x16x4) + argmin + gather + MSE.
// Block = 256 threads = 8 waves; each wave owns 16 rows; block owns 128 rows.
// embed is staged transposed (codeword-major) by async DMA into a
// double-buffered LDS tile: stage st+1 streams in (ASYNCcnt) while stage st
// feeds back-to-back v_wmma_f32_16x16x4_f32 via aligned ds_load_b64 fragments.
// ---------------------------------------------------------------------------
__global__ __launch_bounds__(256) void vq_main_kernel(
    const float* __restrict__ x, const float* __restrict__ embed,
    const float* __restrict__ enorm, float* __restrict__ out) {
  __shared__ float s_emb[2][64 * LDSP]; // 2 x (64 codewords x 64 dims), 34 KB
  __shared__ int   s_idx[128];          // winning codeword per row in block

  const int lane = threadIdx.x & 31;
  const int wave = threadIdx.x >> 5;
  const int m    = lane & 15;           // A: row index / B,C,D: column index
  const int half = lane >> 4;           // K-half selector for A/B fragments
  const long rowBase = ((long)blockIdx.x * 8 + wave) * 16;

#if HAVE_ASYNC_LDS
  // Async DMA transpose: global side coalesced (lanes -> consecutive cols),
  // LDS side scatters to codeword-major. 16 b32 copies per thread per stage.
  auto issue_stage = [&](int st, int buf) {
#pragma unroll
    for (int i = 0; i < 16; ++i) {
      const int e   = threadIdx.x + 256 * i;   // 0..4095
      const int col = e & 63;
      const int d   = e >> 6;
      const float* g = embed + (long)d * NEMBED + st * 64 + col;
      float*       l = &s_emb[buf][col * LDSP + d];
      __builtin_amdgcn_global_load_async_to_lds_b32((gas_i*)g, (las_i*)l, 0, 0);
    }
  };
#endif

  // --- Preload A fragments: a[k] covers K = [4k, 4k+4). ISA 32-bit A layout:
  //     lane(L%16=M): VGPR0 = A[M, 4k+2*half], VGPR1 = A[M, 4k+2*half+1].
  v2f a[16];
  {
    const float* xrow = x + (rowBase + m) * DIM + 2 * half;
#pragma unroll
    for (int k = 0; k < 16; ++k)
      a[k] = *(const v2f*)(xrow + 4 * k);
  }

  float bestScore[8];
  int   bestIdx[8];
#pragma unroll
  for (int v = 0; v < 8; ++v) { bestScore[v] = 3.4e38f; bestIdx[v] = 0; }

#if HAVE_ASYNC_LDS
  issue_stage(0, 0);                    // pipeline prologue
#endif

  // --- Sweep all 2048 codewords in 32 double-buffered stages of 64 columns.
  for (int st = 0; st < NSTAGES; ++st) {
    const int p = st & 1;

#if HAVE_ASYNC_LDS
    if (st + 1 < NSTAGES) {
      // Buffer 1-p was last read in stage st-1; the trailing barrier of that
      // iteration proves every wave is done with it -> safe to overwrite.
      issue_stage(st + 1, 1 - p);
      // Async loads complete in order: counter <= 16 means the 16 copies of
      // stage st are done while stage st+1's 16 are still in flight.
      asm volatile("s_wait_asynccnt 0x10" ::: "memory");
    } else {
      asm volatile("s_wait_asynccnt 0x0" ::: "memory");
    }
    __syncthreads();                    // all waves' stage-st copies visible
#else
    __syncthreads();                    // previous stage fully consumed
#pragma unroll
    for (int t = threadIdx.x; t < 64 * 16; t += 256) {
      const int d  = t >> 4;
      const int c4 = (t & 15) << 2;
      const float4 val = *(const float4*)(embed + (long)d * NEMBED + st * 64 + c4);
      s_emb[0][(c4 + 0) * LDSP + d] = val.x;
      s_emb[0][(c4 + 1) * LDSP + d] = val.y;
      s_emb[0][(c4 + 2) * LDSP + d] = val.z;
      s_emb[0][(c4 + 3) * LDSP + d] = val.w;
    }
    __syncthreads();
#endif

    const float* sbase = &s_emb[HAVE_ASYNC_LDS ? p : 0][0];

#pragma unroll
    for (int ch = 0; ch < 4; ++ch) {
      const int col  = (st * 4 + ch) * 16 + m;  // global codeword for this lane
      const int lcol = ch * 16 + m;             // column within LDS stage

      // B fragments (4x16): lane(N=L%16): VGPR0=B[4k+2*half,N], VGPR1=B[4k+2*half+1,N]
      // = two consecutive floats in codeword-major LDS -> single ds_load_b64.
      const float* bbase = sbase + lcol * LDSP + 2 * half;
      v2f b[16];
#pragma unroll
      for (int k = 0; k < 16; ++k)
        b[k] = *(const v2f*)(bbase + 4 * k);

      v8f acc = {};
#pragma unroll
      for (int k = 0; k < 16; ++k) {
        acc = __builtin_amdgcn_wmma_f32_16x16x4_f32(
            /*neg_a=*/false, a[k], /*neg_b=*/false, b[k],
            /*c_mod=*/(short)0, acc, /*reuse_a=*/false, /*reuse_b=*/false);
      }

      // score = ||e||^2 - 2*(f.e)  (row-constant ||f||^2 dropped: argmin-invariant)
      const float en = enorm[col];
#pragma unroll
      for (int v = 0; v < 8; ++v) {
        const float sc = fmaf(-2.0f, acc[v], en);
        if (sc < bestScore[v]) { bestScore[v] = sc; bestIdx[v] = col; }
      }
    }

#if HAVE_ASYNC_LDS
    __syncthreads();  // all waves done reading buffer p -> next iter may refill
#endif
  }

  // --- Min-reduce across the 16 lanes of each half-group (wave32 shuffles).
  //     Slot v holds row (v + 8*half); candidates spread over lane%16.
#pragma unroll
  for (int off = 1; off < 16; off <<= 1) {
#pragma unroll
    for (int v = 0; v < 8; ++v) {
      const float os = __shfl_xor(bestScore[v], off, 32);
      const int   oi = __shfl_xor(bestIdx[v],  off, 32);
      if (os < bestScore[v] || (os == bestScore[v] && oi < bestIdx[v])) {
        bestScore[v] = os; bestIdx[v] = oi;
      }
    }
  }
  if (m == 0) {  // lanes 0 and 16 hold rows 0..7 and 8..15 respectively
#pragma unroll
    for (int v = 0; v < 8; ++v)
      s_idx[wave * 16 + half * 8 + v] = bestIdx[v];
  }
  __syncthreads();

  // --- Output pass: quantize gather (fp32), indices, diff partial sum.
  float* __restrict__ qout   = out;
  float* __restrict__ diff   = out + DIFF_OFF;
  float* __restrict__ indout = out + IND_OFF;
  const long blockRow = (long)blockIdx.x * 128;

  if (threadIdx.x < 128)
    indout[blockRow + threadIdx.x] = (float)s_idx[threadIdx.x];

  float psum = 0.0f;
#pragma unroll
  for (int i = 0; i < 32; ++i) {
    const int e   = threadIdx.x + 256 * i;       // 0..8191, coalesced
    const int row = e >> 6;
    const int d   = e & 63;
    const int ind = s_idx[row];
    const float q  = embed[(long)d * NEMBED + ind];  // gather, L2-resident
    const float xv = x[blockRow * DIM + e];
    const float dd = q - xv;
    psum += dd * dd;
    qout[blockRow * DIM + e] = q;
  }
#pragma unroll
  for (int off = 16; off >= 1; off >>= 1)
    psum += __shfl_xor(psum, off, 32);
  if (lane == 0)
    atomicAdd(diff, psum * (1.0f / (float)NTOTAL));
}

// ---------------------------------------------------------------------------
extern "C" void kernel_launch(void* const* d_in, const int* in_sizes, int n_in,
                              void* d_out, int out_size, void* d_ws, size_t ws_size,
                              hipStream_t stream) {
  const float* x     = (const float*)d_in[0];   // [16,64,64,64] fp32
  const float* embed = (const float*)d_in[1];   // [64,2048] fp32
  float* out   = (float*)d_out;                 // [quantize | diff | ind]
  float* enorm = (float*)d_ws;                  // 2048 floats scratch

  vq_enorm_kernel<<<(NEMBED + 255) / 256, 256, 0, stream>>>(embed, enorm,
                                                            out + DIFF_OFF);
  vq_main_kernel<<<NROWS / 128, 256, 0, stream>>>(x, embed, enorm, out);
}